// MultiplySparsemax_10720238370934
// MI455X (gfx1250) — compile-verified
//
#include <hip/hip_runtime.h>
#include <hip/hip_bf16.h>

// Problem constants (fixed by the reference: (16, 11, 65536) f32, LST=64).
#define NI      11          // instruments (sparsemax axis 1)
#define TIME    65536       // time length (divisible by 64 -> no padding)
#define LSTLEN  64          // sparsemax time-frame length
#define TCHUNK  256         // time positions per block = 4 frames
#define NFRM    (TCHUNK / LSTLEN)   // 4
#define NGRP    (NI * NFRM)         // 44 (instrument, frame) groups per block
#define NTHREADS 256
#define NV4     (NI * (TCHUNK / 4)) // 704 float4 tiles to stage per block

// Typed address-space pointers for the gfx1250 async global->LDS builtins.
// b32 variant takes AS1/AS3 int*; b128 variant takes AS1/AS3 v4i*
// (diagnostic-confirmed: '__vector_size__(4 * sizeof(int)) int *').
typedef int v4i __attribute__((__vector_size__(16)));
typedef __attribute__((address_space(1))) int  gl_int_t;
typedef __attribute__((address_space(3))) int  lds_int_t;
typedef __attribute__((address_space(1))) v4i  gl_v4i_t;
typedef __attribute__((address_space(3))) v4i  lds_v4i_t;

// ---- wave32 cross-lane f32 add-reduction (5 shuffle steps) ----
__device__ __forceinline__ float wred_add(float v) {
  v += __shfl_xor(v, 16, 32);
  v += __shfl_xor(v, 8, 32);
  v += __shfl_xor(v, 4, 32);
  v += __shfl_xor(v, 2, 32);
  v += __shfl_xor(v, 1, 32);
  return v;
}

// Exact sparsemax threshold over 64 values held 2-per-lane across one wave.
// Michelot fixpoint: tau = (sum_{z>tau} z - 1)/count; the support shrinks
// monotonically and the iteration terminates when the support count is
// stable (<= 64 iters, typically ~4). The support count K is computed with
// ballot+popc (wave-uniform immediately, no shuffle tree); only the sum S
// needs the 5-step shuffle reduction.
__device__ __forceinline__ float tau_sparsemax64(float a, float b) {
  float tau = (wred_add(a + b) - 1.0f) * (1.0f / 64.0f);
  int kprev = 64;
#pragma unroll 1
  for (int it = 0; it < 64; ++it) {
    bool ma = a > tau;
    bool mb = b > tau;
    int K = __popc(__builtin_amdgcn_ballot_w32(ma)) +
            __popc(__builtin_amdgcn_ballot_w32(mb));
    float S = wred_add((ma ? a : 0.0f) + (mb ? b : 0.0f));
    if (K == kprev) break;      // support unchanged -> tau is the fixpoint
    tau = (S - 1.0f) / (float)K;
    kprev = K;
  }
  return tau;
}

// Exact sparsemax threshold over 11 register-resident values (per thread).
// Per-lane Michelot; lanes that converged hold tau fixed, wave exits when
// every lane's support is stable.
__device__ __forceinline__ float tau_sparsemax11(const float z[NI]) {
  float s = 0.0f;
#pragma unroll
  for (int i = 0; i < NI; ++i) s += z[i];
  float tau = (s - 1.0f) * (1.0f / (float)NI);
  float kprev = (float)NI;
#pragma unroll 1
  for (int it = 0; it < NI; ++it) {
    float S = 0.0f, K = 0.0f;
#pragma unroll
    for (int i = 0; i < NI; ++i) {
      bool m = z[i] > tau;
      S += m ? z[i] : 0.0f;
      K += m ? 1.0f : 0.0f;
    }
    bool done = (K == kprev);
    if (!done) { tau = (S - 1.0f) / K; kprev = K; }
    if (__builtin_amdgcn_ballot_w32(!done) == 0u) break;
  }
  return tau;
}

__global__ __launch_bounds__(NTHREADS, 4) void
sparsemax_mul_kernel(const float* __restrict__ in, float* __restrict__ out) {
  // No row padding: all LDS accesses below are stride-1 across lanes, and
  // unpadded rows keep every row start (1 KB stride) 16-byte aligned for
  // the b128 async copies.
  __shared__ float tile[NI][TCHUNK];       // 11 KB
  __shared__ float tauT[NGRP];             // per-(instrument,frame) time tau

  const int t      = threadIdx.x;              // time position in chunk
  const int b      = blockIdx.x >> 8;          // 256 chunks per batch
  const int chunk  = blockIdx.x & 255;
  const size_t base = (size_t)b * NI * TIME + (size_t)chunk * TCHUNK;

  // ---- Phase A: stage the 11 x 256 tile into LDS (gfx1250 async copies,
  // ASYNCcnt tracked). Preferred: 16B/lane b128 transfers, 704 total,
  // striped flat across the 256 threads. ----
#if defined(__gfx1250__) && __has_builtin(__builtin_amdgcn_global_load_async_to_lds_b128)
#pragma unroll
  for (int w = 0; w < (NV4 + NTHREADS - 1) / NTHREADS; ++w) {
    const int f = t + w * NTHREADS;
    if (f < NV4) {
      const int i = f >> 6;            // instrument row
      const int c = (f & 63) << 2;     // float offset within row (16B aligned)
      __builtin_amdgcn_global_load_async_to_lds_b128(
          (gl_v4i_t*)(in + base + (size_t)i * TIME + c),
          (lds_v4i_t*)&tile[i][c],
          /*offset=*/0, /*cpol=*/0);
    }
  }
#if __has_builtin(__builtin_amdgcn_s_wait_asynccnt)
  __builtin_amdgcn_s_wait_asynccnt(0);
#else
  asm volatile("s_wait_asynccnt 0x0" ::: "memory");
#endif
#elif defined(__gfx1250__) && __has_builtin(__builtin_amdgcn_global_load_async_to_lds_b32)
#pragma unroll
  for (int i = 0; i < NI; ++i) {
    __builtin_amdgcn_global_load_async_to_lds_b32(
        (gl_int_t*)(in + base + (size_t)i * TIME + t),
        (lds_int_t*)&tile[i][t],
        /*offset=*/0, /*cpol=*/0);
  }
#if __has_builtin(__builtin_amdgcn_s_wait_asynccnt)
  __builtin_amdgcn_s_wait_asynccnt(0);
#else
  asm volatile("s_wait_asynccnt 0x0" ::: "memory");
#endif
#else
  // Fallback: plain coalesced loads (single-use data -> non-temporal).
#pragma unroll
  for (int i = 0; i < NI; ++i)
    tile[i][t] = __builtin_nontemporal_load(in + base + (size_t)i * TIME + t);
#endif
  __syncthreads();

  // ---- Phase B: time-axis sparsemax taus, one wave per 64-frame group ----
  const int wave = t >> 5;
  const int lane = t & 31;
  for (int g = wave; g < NGRP; g += (NTHREADS / 32)) {
    const int i  = g >> 2;      // instrument
    const int sb = g & 3;       // frame within chunk
    float a  = tile[i][sb * LSTLEN + lane];
    float bb = tile[i][sb * LSTLEN + 32 + lane];
    float tau = tau_sparsemax64(a, bb);
    if (lane == 0) tauT[g] = tau;
  }
  __syncthreads();

  // ---- Phase C: instrument-axis sparsemax per time position + product ----
  float z[NI];
#pragma unroll
  for (int i = 0; i < NI; ++i) z[i] = tile[i][t];

  const float tau_i = tau_sparsemax11(z);
  const int   sb    = t >> 6;

#pragma unroll
  for (int i = 0; i < NI; ++i) {
    float vi = z[i] - tau_i;            vi = (vi > 0.0f) ? vi : 0.0f;
    float vt = z[i] - tauT[i * 4 + sb]; vt = (vt > 0.0f) ? vt : 0.0f;
    // Output is write-once, never re-read: stream past the caches.
    __builtin_nontemporal_store(vi * vt, out + base + (size_t)i * TIME + t);
  }
}

extern "C" void kernel_launch(void* const* d_in, const int* in_sizes, int n_in,
                              void* d_out, int out_size, void* d_ws, size_t ws_size,
                              hipStream_t stream) {
  (void)n_in; (void)out_size; (void)d_ws; (void)ws_size;
  const float* in  = (const float*)d_in[0];
  float*       out = (float*)d_out;

  const int total = in_sizes[0];               // 16 * 11 * 65536
  const int batch = total / (NI * TIME);       // 16
  dim3 grid(batch * (TIME / TCHUNK));          // 16 * 256 = 4096 blocks
  dim3 block(NTHREADS);
  sparsemax_mul_kernel<<<grid, block, 0, stream>>>(in, out);
}